// Connectome_5368709120658
// MI455X (gfx1250) — compile-verified
//
#include <hip/hip_runtime.h>
#include <hip/hip_bf16.h>
#include <stdint.h>

// ---------------------------------------------------------------------------
// Problem constants (shapes are static in the reference)
// ---------------------------------------------------------------------------
#define NB   16      // batch
#define TT   200     // timepoints (multiple of 4 -> clean K loop for wmma)
#define HH   144
#define WW   320
#define VV   (HH * WW)        // 46080 pixels
#define RR   200              // ROIs (segment 0 = background, dropped)
#define RP   208              // ROIs padded to 13 wmma tiles of 16
#define NTILE 13              // RP / 16
#define PAIRS 19900           // RR*(RR-1)/2
#define EPSN 1e-8f

typedef float v2f __attribute__((ext_vector_type(2)));
typedef float v8f __attribute__((ext_vector_type(8)));

// ---------------------------------------------------------------------------
// Kernel A: fold mask into parcellation (seg in 0..200, u8) + 1/count per ROI
// Single block; LDS integer atomics.
// NOTE: jax bool arrays are 1 byte/elem -> read mask as uint8.
// ---------------------------------------------------------------------------
__global__ void seg_kernel(const int* __restrict__ parc,
                           const uint8_t* __restrict__ mask,
                           uint8_t* __restrict__ seg,
                           float* __restrict__ inv) {
  __shared__ int cnt[256];
  const int tid = threadIdx.x;
  for (int i = tid; i < 256; i += blockDim.x) cnt[i] = 0;
  __syncthreads();
  for (int v = tid; v < VV; v += blockDim.x) {
    int s = mask[v] ? parc[v] : 0;
    seg[v] = (uint8_t)s;
    if (s > 0) atomicAdd(&cnt[s - 1], 1);
  }
  __syncthreads();
  for (int r = tid; r < RR; r += blockDim.x) {
    int c = cnt[r];
    inv[r] = 1.0f / (float)(c > 0 ? c : 1);
  }
}

// ---------------------------------------------------------------------------
// Kernel B: segment means. One 256-thread block per (n,t).
// Streams 184KB of x coalesced; seg (46KB) stays L2-resident.
// LDS float atomics into 200 bins (background skipped).
// Writes roi into Z buffer laid out (NB, TT, RP) row-major.
// ---------------------------------------------------------------------------
__global__ void roi_kernel(const float* __restrict__ x,
                           const uint8_t* __restrict__ seg,
                           const float* __restrict__ inv,
                           float* __restrict__ z) {
  __shared__ float bins[256];
  const int nt  = blockIdx.x;            // 0 .. NB*TT-1
  const int tid = threadIdx.x;
  for (int i = tid; i < 256; i += blockDim.x) bins[i] = 0.0f;
  __syncthreads();
  const float* xr = x + (size_t)nt * VV;
  for (int v = tid; v < VV; v += blockDim.x) {
    int s = seg[v];
    if (s > 0) atomicAdd(&bins[s - 1], xr[v]);
  }
  __syncthreads();
  float* zrow = z + (size_t)nt * RP;
  for (int r = tid; r < RR; r += blockDim.x)
    zrow[r] = bins[r] * inv[r];
}

// ---------------------------------------------------------------------------
// Kernel C: per (n, r) column over time: demean, scale by 1/(||.||2 + eps).
// Pad columns (r >= RR) are written as zeros so they are inert in the Gram.
// ---------------------------------------------------------------------------
__global__ void norm_kernel(float* __restrict__ z) {
  const int idx = blockIdx.x * blockDim.x + threadIdx.x;
  if (idx >= NB * RP) return;
  const int n = idx / RP;
  const int r = idx - n * RP;
  float* col = z + (size_t)n * TT * RP + r;
  if (r >= RR) {
    for (int t = 0; t < TT; ++t) col[(size_t)t * RP] = 0.0f;
    return;
  }
  float mean = 0.0f;
  for (int t = 0; t < TT; ++t) mean += col[(size_t)t * RP];
  mean *= (1.0f / (float)TT);
  float ss = 0.0f;
  for (int t = 0; t < TT; ++t) {
    float d = col[(size_t)t * RP] - mean;
    ss += d * d;
  }
  const float sc = 1.0f / (sqrtf(ss) + EPSN);
  for (int t = 0; t < TT; ++t)
    col[(size_t)t * RP] = (col[(size_t)t * RP] - mean) * sc;
}

// ---------------------------------------------------------------------------
// Kernel D: conn[n] = Z^T Z via V_WMMA_F32_16X16X4_F32 (full f32 WMMA).
// One wave32 per 16x16 tile; only upper-triangular tiles do work.
// A-frag (16x4 f32, 2 VGPRs): lanes 0-15 carry K=0,1 ; lanes 16-31 carry K=2,3
// for M = lane&15. B-frag (4x16) mirrors with N = lane&15.
// D layout: lanes 0-15 -> M=g, N=lane ; lanes 16-31 -> M=g+8, N=lane-16.
// Each lane scatters its 8 results straight into the triu-flattened output.
// ---------------------------------------------------------------------------
__global__ void gram_kernel(const float* __restrict__ z,
                            float* __restrict__ out) {
  const int tr = blockIdx.x;   // row tile   0..12
  const int ts = blockIdx.y;   // col tile   0..12
  const int n  = blockIdx.z;   // batch      0..15
  if (ts < tr) return;         // strictly-lower tiles contribute nothing

  const int lane = threadIdx.x;       // 0..31, wave32, EXEC all ones
  const int half = lane >> 4;         // 0: K=0,1   1: K=2,3
  const int l    = lane & 15;

  const float* zn = z + (size_t)n * TT * RP;
  const int r0 = tr * 16;
  const int s0 = ts * 16;

  v8f c = {0.f, 0.f, 0.f, 0.f, 0.f, 0.f, 0.f, 0.f};

  for (int k0 = 0; k0 < TT; k0 += 4) {
    const int ka = k0 + (half ? 2 : 0);
    v2f a, b;
    a.x = zn[(size_t)(ka)     * RP + r0 + l];
    a.y = zn[(size_t)(ka + 1) * RP + r0 + l];
    b.x = zn[(size_t)(ka)     * RP + s0 + l];
    b.y = zn[(size_t)(ka + 1) * RP + s0 + l];
    // 8 args: (neg_a, A, neg_b, B, c_mod, C, reuse_a, reuse_b)
    c = __builtin_amdgcn_wmma_f32_16x16x4_f32(false, a, false, b,
                                              (short)0, c, false, false);
  }

  float* outn = out + (size_t)n * PAIRS;
#pragma unroll
  for (int g = 0; g < 8; ++g) {
    const int m = half ? (g + 8) : g;
    const int r = r0 + m;
    const int s = s0 + l;
    if (r < RR && s < RR && r < s) {
      const int idx = r * (2 * RR - r - 1) / 2 + (s - r - 1);
      outn[idx] = c[g];
    }
  }
}

// ---------------------------------------------------------------------------
// Launcher
// ---------------------------------------------------------------------------
extern "C" void kernel_launch(void* const* d_in, const int* in_sizes, int n_in,
                              void* d_out, int out_size, void* d_ws, size_t ws_size,
                              hipStream_t stream) {
  const float*   x    = (const float*)d_in[0];     // (16,1,200,144,320) f32
  const int*     parc = (const int*)d_in[1];       // (144,320) i32
  const uint8_t* mask = (const uint8_t*)d_in[2];   // (144,320) bool (1B)
  float*         out  = (float*)d_out;             // (16, 19900) f32

  uint8_t* ws  = (uint8_t*)d_ws;
  uint8_t* seg = ws;                                 // VV bytes (46080, 256-aligned)
  float*   inv = (float*)(ws + VV);                  // 200 f32
  float*   z   = (float*)(ws + VV + 1024);           // NB*TT*RP f32 = 2.66 MB

  seg_kernel<<<1, 1024, 0, stream>>>(parc, mask, seg, inv);

  roi_kernel<<<NB * TT, 256, 0, stream>>>(x, seg, inv, z);

  {
    const int threads = NB * RP;            // 3328
    norm_kernel<<<(threads + 255) / 256, 256, 0, stream>>>(z);
  }

  gram_kernel<<<dim3(NTILE, NTILE, NB), 32, 0, stream>>>(z, out);
}